// GCNConv_51084341018871
// MI455X (gfx1250) — compile-verified
//
#include <hip/hip_runtime.h>

typedef __attribute__((ext_vector_type(2))) float v2f;
typedef __attribute__((ext_vector_type(8))) float v8f;

#define INCH   128
#define OUTCH  128
#define PSTRIDE 144   // pair-row stride in float2 units: 2*144 % 64 == 32 words
                      // -> the two lane-halves read disjoint LDS banks

// -------------------------------------------------------------------------
// Kernel 1: out[n][c] = bias[c]  (folds the "+ bias" and zero-init together)
// -------------------------------------------------------------------------
__global__ __launch_bounds__(256) void gcn_init_bias(
    const float* __restrict__ bias, float* __restrict__ out, long long total4)
{
    long long i = (long long)blockIdx.x * 256 + threadIdx.x;   // float4 units
    if (i >= total4) return;
    int c4 = (int)(i & (OUTCH / 4 - 1));
    ((float4*)out)[i] = ((const float4*)bias)[c4];
}

// -------------------------------------------------------------------------
// Kernel 2: h = x @ W via V_WMMA_F32_16X16X4_F32 (full f32 precision).
// Block = 256 threads = 8 waves; wave w owns rows [blk*128 + w*16, +16),
// all 128 output columns (8 accumulator tiles of 16x16).
// W is staged in LDS as K-pairs (float2 {W[k][n], W[k+1][n]}) so every
// B fragment is one ds_load_b64 straight into an aligned VGPR pair.
// -------------------------------------------------------------------------
__global__ __launch_bounds__(256) void gcn_gemm_wmma(
    const float* __restrict__ x, const float* __restrict__ w,
    float* __restrict__ h, int n_nodes)
{
    __shared__ float2 lw2[(INCH / 2) * PSTRIDE];   // 64 pair-rows x 144 -> 72KB

    const int tid = threadIdx.x;

    // Stage W: pair-row p holds {W[2p][n], W[2p+1][n]} for n = 0..127.
    for (int i = tid; i < (INCH / 2) * OUTCH; i += 256) {
        int p = i >> 7;          // pair-row
        int n = i & 127;         // column
        float2 f;
        f.x = w[(2 * p    ) * OUTCH + n];
        f.y = w[(2 * p + 1) * OUTCH + n];
        lw2[p * PSTRIDE + n] = f;
    }
    __syncthreads();

    const int wave = tid >> 5;
    const int lane = tid & 31;
    const int half = lane >> 4;    // 0: lanes 0-15, 1: lanes 16-31
    const int l16  = lane & 15;

    const long long m0 = ((long long)blockIdx.x * 8 + wave) * 16;
    if (m0 >= n_nodes) return;     // wave-uniform: EXEC stays all-ones for WMMA

    const bool full = (m0 + 16 <= (long long)n_nodes);   // wave-uniform

    // A-fragment source row (clamped only on the ragged tail).
    long long mrow = m0 + l16;
    if (!full && mrow >= n_nodes) mrow = n_nodes - 1;
    // A 16x4 f32 layout: lanes 0-15 hold K=k0+0,k0+1 ; lanes 16-31 hold K=k0+2,k0+3
    const float* xrow = x + mrow * INCH + half * 2;

    v8f acc[8];
    const v8f vzero = {0.f, 0.f, 0.f, 0.f, 0.f, 0.f, 0.f, 0.f};
#pragma unroll
    for (int t = 0; t < 8; ++t) acc[t] = vzero;

#pragma unroll 2
    for (int k0 = 0; k0 < INCH; k0 += 4) {
        float2 av = *(const float2*)(xrow + k0);   // global_load_b64
        v2f a; a.x = av.x; a.y = av.y;

        // B 4x16 f32 layout (mirror of A): lane half selects K pair, N = l16.
        const float2* wb = &lw2[((k0 >> 1) + half) * PSTRIDE + l16];
#pragma unroll
        for (int t = 0; t < 8; ++t) {
            v2f b = *(const v2f*)&wb[t * 16];      // one ds_load_b64
            acc[t] = __builtin_amdgcn_wmma_f32_16x16x4_f32(
                false, a, false, b, (short)0, acc[t], false, false);
        }
    }

    // D 16x16 f32 layout: VGPR j -> row (j + half*8), N = l16. Coalesced
    // 64B row-segment stores per lane half.
    float* hbase = h + m0 * OUTCH + l16;
    if (full) {
#pragma unroll
        for (int t = 0; t < 8; ++t) {
#pragma unroll
            for (int j = 0; j < 8; ++j) {
                hbase[(j + half * 8) * OUTCH + t * 16] = acc[t][j];
            }
        }
    } else {
#pragma unroll
        for (int t = 0; t < 8; ++t) {
#pragma unroll
            for (int j = 0; j < 8; ++j) {
                if (m0 + j + half * 8 < n_nodes)
                    hbase[(j + half * 8) * OUTCH + t * 16] = acc[t][j];
            }
        }
    }
}

// -------------------------------------------------------------------------
// Kernel 3: edge scatter. One wave32 per edge; float4 per lane covers all
// 128 channels. h and out are L2-resident (51.2MB each vs 192MB L2), so the
// random gather + FP32 global atomics stay on-die.
// -------------------------------------------------------------------------
__global__ __launch_bounds__(256) void gcn_spmm_scatter(
    const float* __restrict__ h,
    const int* __restrict__ erow, const int* __restrict__ ecol,
    const float* __restrict__ ew, float* __restrict__ out, int n_edges)
{
    const int wave = threadIdx.x >> 5;
    const int lane = threadIdx.x & 31;
    const int e = blockIdx.x * 8 + wave;     // wave-uniform edge id
    if (e >= n_edges) return;

    const int   r   = erow[e];
    const int   c   = ecol[e];
    const float wgt = ew[e];

    const float4 v = ((const float4*)(h + (long long)c * OUTCH))[lane];
    float* dst = out + (long long)r * OUTCH + lane * 4;

    // Hardware GLOBAL_ATOMIC_ADD_F32 (no CAS loop).
    unsafeAtomicAdd(dst + 0, wgt * v.x);
    unsafeAtomicAdd(dst + 1, wgt * v.y);
    unsafeAtomicAdd(dst + 2, wgt * v.z);
    unsafeAtomicAdd(dst + 3, wgt * v.w);
}

// -------------------------------------------------------------------------
extern "C" void kernel_launch(void* const* d_in, const int* in_sizes, int n_in,
                              void* d_out, int out_size, void* d_ws, size_t ws_size,
                              hipStream_t stream)
{
    const float* x    = (const float*)d_in[0];
    const float* w    = (const float*)d_in[1];
    const float* bias = (const float*)d_in[2];
    const int*   erow = (const int*)d_in[3];
    const int*   ecol = (const int*)d_in[4];
    const float* ew   = (const float*)d_in[5];
    float* out = (float*)d_out;
    float* h   = (float*)d_ws;    // needs n_nodes * 128 * 4 = 51.2 MB scratch

    const int n_nodes = in_sizes[0] / INCH;
    const int n_edges = in_sizes[3];

    // 1) out = broadcast(bias)
    long long total4 = (long long)n_nodes * (OUTCH / 4);
    int init_blocks = (int)((total4 + 255) / 256);
    hipLaunchKernelGGL(gcn_init_bias, dim3(init_blocks), dim3(256), 0, stream,
                       bias, out, total4);

    // 2) h = x @ W  (WMMA f32)
    int tiles = (n_nodes + 15) / 16;
    int gemm_blocks = (tiles + 7) / 8;
    hipLaunchKernelGGL(gcn_gemm_wmma, dim3(gemm_blocks), dim3(256), 0, stream,
                       x, w, h, n_nodes);

    // 3) out[row] += w_e * h[col]
    int spmm_blocks = (n_edges + 7) / 8;
    hipLaunchKernelGGL(gcn_spmm_scatter, dim3(spmm_blocks), dim3(256), 0, stream,
                       h, erow, ecol, ew, out, n_edges);
}